// Psi11t_25640954757286
// MI455X (gfx1250) — compile-verified
//
#include <hip/hip_runtime.h>
#include <stdint.h>

// Lattice / tiling parameters
#define LX 512
#define LY 512
#define TX 32               // tile extent along x (axis 2)
#define TY 64               // tile extent along y (axis 3, contiguous)
#define HX (TX + 4)         // s halo-2 tile: 36
#define HY (TY + 4)         // 68
#define BXD (TX + 2)        // b halo-1 tile: 34
#define BYD (TY + 2)        // 66
#define NTHREADS 256

#if __has_builtin(__builtin_amdgcn_global_load_async_to_lds_b32)
#define USE_ASYNC_BUILTIN 1
#endif

typedef __attribute__((address_space(3))) int lds_int;
typedef __attribute__((address_space(1))) int glb_int;

// Generic->LDS: low 32 bits of a generic LDS pointer are the LDS byte offset
// (ISA 10.2 aperture rules), so an integer round-trip is a valid addrspace cast.
__device__ __forceinline__ lds_int* to_lds(void* p) {
    return (lds_int*)(uint32_t)(uintptr_t)p;
}
__device__ __forceinline__ glb_int* to_glb(const void* p) {
    return (glb_int*)(uintptr_t)p;
}

// Async copy of one DWORD from global memory into LDS (per-lane addresses),
// tracked by ASYNCcnt (CDNA5 ISA).
__device__ __forceinline__ void async_copy_b32(const float* gsrc, float* ldst) {
#if defined(USE_ASYNC_BUILTIN)
    __builtin_amdgcn_global_load_async_to_lds_b32(to_glb(gsrc), to_lds(ldst), 0, 0);
#else
    uint32_t loff = (uint32_t)(uintptr_t)ldst;   // LDS byte address (VDST)
    uint64_t gaddr = (uint64_t)(uintptr_t)gsrc;  // 64-bit global address (VADDR)
    asm volatile("global_load_async_to_lds_b32 %0, %1, off"
                 :
                 : "v"(loff), "v"(gaddr)
                 : "memory");
#endif
}

__device__ __forceinline__ void wait_async_zero() {
#if __has_builtin(__builtin_amdgcn_s_wait_asynccnt)
    __builtin_amdgcn_s_wait_asynccnt(0);
#else
    asm volatile("s_wait_asynccnt 0x0" ::: "memory");
#endif
}

// out[b,a,x,y] = 2*t*c0 * ( s x F )_a
//   with b = s . (sum of 4 nbrs of s),  Fs = sum of 4 nbrs of s,
//        Fbs = sum of 4 nbrs of (s*b),  F = Fs*b_site + Fbs
__global__ __launch_bounds__(NTHREADS)
void psi11t_force_kernel(const float* __restrict__ s,
                         const float* __restrict__ t,
                         const float* __restrict__ c0,
                         float* __restrict__ out) {
    __shared__ float sh_s[3][HX * HY];   // 3 * 36*68 * 4B = 29376 B
    __shared__ float sh_b[BXD * BYD];    // 34*66 * 4B = 8976 B

    const int tid = threadIdx.x;
    const int y0 = blockIdx.x * TY;
    const int x0 = blockIdx.y * TX;
    const int bb = blockIdx.z;

    const size_t plane = (size_t)LX * LY;
    const float* __restrict__ sb = s + (size_t)bb * 3 * plane;
    float* __restrict__ ob = out + (size_t)bb * 3 * plane;

    // ---- Phase 1: async-load (HX x HY) x 3 halo tile of s into LDS ----
    for (int i = tid; i < HX * HY; i += NTHREADS) {
        int tx = i / HY;
        int ty = i - tx * HY;
        int gx = (x0 + tx - 2) & (LX - 1);   // periodic wrap
        int gy = (y0 + ty - 2) & (LY - 1);
        size_t gidx = (size_t)gx * LY + gy;
        async_copy_b32(sb + gidx,              &sh_s[0][i]);
        async_copy_b32(sb + plane + gidx,      &sh_s[1][i]);
        async_copy_b32(sb + 2 * plane + gidx,  &sh_s[2][i]);
    }
    wait_async_zero();
    __syncthreads();

    // ---- Phase 2: b = s . (sum of 4 neighbors) on core+halo1 region ----
    for (int i = tid; i < BXD * BYD; i += NTHREADS) {
        int tx = i / BYD;
        int ty = i - tx * BYD;
        int sx = tx + 1, sy = ty + 1;        // position within sh_s coords
        float acc = 0.0f;
#pragma unroll
        for (int c = 0; c < 3; ++c) {
            float v = sh_s[c][sx * HY + sy];
            float n = sh_s[c][(sx + 1) * HY + sy] + sh_s[c][(sx - 1) * HY + sy] +
                      sh_s[c][sx * HY + sy + 1] + sh_s[c][sx * HY + sy - 1];
            acc = fmaf(v, n, acc);
        }
        sh_b[i] = acc;
    }
    __syncthreads();

    const float coeff = 2.0f * t[0] * c0[0];

    // ---- Phase 3: F and cross product on the core tile ----
    for (int i = tid; i < TX * TY; i += NTHREADS) {
        int tx = i / TY;
        int ty = i - tx * TY;
        int cx = tx + 2, cy = ty + 2;        // sh_s coords
        int bx = tx + 1, by = ty + 1;        // sh_b coords

        float bsite = sh_b[bx * BYD + by];
        float bxp = sh_b[(bx + 1) * BYD + by];
        float bxm = sh_b[(bx - 1) * BYD + by];
        float byp = sh_b[bx * BYD + by + 1];
        float bym = sh_b[bx * BYD + by - 1];

        float sv[3], Fv[3];
#pragma unroll
        for (int c = 0; c < 3; ++c) {
            float sxp = sh_s[c][(cx + 1) * HY + cy];
            float sxm = sh_s[c][(cx - 1) * HY + cy];
            float syp = sh_s[c][cx * HY + cy + 1];
            float sym = sh_s[c][cx * HY + cy - 1];
            sv[c] = sh_s[c][cx * HY + cy];
            float Fs = sxp + sxm + syp + sym;                 // neighbor sum of s
            float Fbs = sxp * bxp + sxm * bxm + syp * byp + sym * bym; // nbr sum of s*b
            Fv[c] = fmaf(Fs, bsite, Fbs);
        }

        size_t gidx = (size_t)(x0 + tx) * LY + (y0 + ty);
        // out = coeff * (s x F); non-temporal stores keep the 96MB input
        // L2-resident (192MB L2) so halo re-reads stay on-chip.
        __builtin_nontemporal_store(coeff * (sv[1] * Fv[2] - sv[2] * Fv[1]), ob + gidx);
        __builtin_nontemporal_store(coeff * (sv[2] * Fv[0] - sv[0] * Fv[2]), ob + plane + gidx);
        __builtin_nontemporal_store(coeff * (sv[0] * Fv[1] - sv[1] * Fv[0]), ob + 2 * plane + gidx);
    }
}

extern "C" void kernel_launch(void* const* d_in, const int* in_sizes, int n_in,
                              void* d_out, int out_size, void* d_ws, size_t ws_size,
                              hipStream_t stream) {
    const float* s  = (const float*)d_in[0];
    const float* t  = (const float*)d_in[1];
    const float* c0 = (const float*)d_in[2];
    float* out = (float*)d_out;

    int B = in_sizes[0] / (3 * LX * LY);   // 32
    dim3 grid(LY / TY, LX / TX, B);        // (8, 16, 32)
    dim3 block(NTHREADS);
    hipLaunchKernelGGL(psi11t_force_kernel, grid, block, 0, stream, s, t, c0, out);
}